// GNN_9079560863943
// MI455X (gfx1250) — compile-verified
//
#include <hip/hip_runtime.h>
#include <math.h>

// ---------------------------------------------------------------------------
// GCN forward for MI455X (gfx1250, wave32).
//   - GEMMs: V_WMMA_F32_16X16X4_F32, A tile staged in LDS (shared by 8 waves),
//     two interleaved accumulators to break the D->C dependency chain.
//   - Edge aggregation: one wave per edge, float4 lanes, global f32 atomics
//     (scatter-bound; gather table is L2-resident: 25.6 MB << 192 MB L2).
// ---------------------------------------------------------------------------

typedef __attribute__((ext_vector_type(2))) float v2f;
typedef __attribute__((ext_vector_type(8))) float v8f;

#define FEATS 128          // IN_FEATS == HIDDEN == 128
#define OUTF  64           // OUT_SIZE
#define NGRAPH 512
#define BN_EPS 1e-5f

__device__ __forceinline__ float mish_f(float x) {
    float sp = (x > 20.0f) ? x : log1pf(__expf(x));   // stable softplus
    return x * tanhf(sp);
}

// ---------------------------------------------------------------------------
// WMMA GEMM:  C[M,N] = A[M,K] @ B[K,N]   (row-major, M%16==0, N%16==0, K==128)
// grid.x = M/16; blockDim.x = (N/16)*32 -> each wave owns one 16x16 tile of
// the block's 16xN tile-row. The 16xK A tile (contiguous in memory) is staged
// once in LDS and shared by all waves.
// ---------------------------------------------------------------------------
__global__ void gemm_wmma_f32(const float* __restrict__ A,
                              const float* __restrict__ Bw,
                              float* __restrict__ C,
                              const float* __restrict__ bias,
                              int M, int K, int N, int fuse_mish) {
    __shared__ float As[16 * FEATS];                    // 8 KB

    const int lane = threadIdx.x & 31;
    const int wave = threadIdx.x >> 5;
    const int half = lane >> 4;        // 0: K pair {0,1}, 1: K pair {2,3}
    const int l15  = lane & 15;
    const int m0   = blockIdx.x * 16;
    const int n0   = wave * 16;

    // cooperative copy of the contiguous 16xK A tile into LDS
    const float* Ag = A + (size_t)m0 * K;
    for (int idx = threadIdx.x * 4; idx < 16 * K; idx += blockDim.x * 4)
        *(float4*)&As[idx] = *(const float4*)&Ag[idx];
    __syncthreads();

    const float* arow = As + l15 * K + half * 2;        // LDS fragment base
    const float* bcol = Bw + n0 + l15;

    v8f acc0 = {}, acc1 = {};
    for (int k0 = 0; k0 < K; k0 += 8) {
        v2f a0 = *(const v2f*)(arow + k0);              // ds_load_b64
        v2f a1 = *(const v2f*)(arow + k0 + 4);
        const int kb = k0 + half * 2;
        v2f b0, b1;
        b0.x = bcol[(size_t)(kb    ) * N];
        b0.y = bcol[(size_t)(kb + 1) * N];
        b1.x = bcol[(size_t)(kb + 4) * N];
        b1.y = bcol[(size_t)(kb + 5) * N];
        acc0 = __builtin_amdgcn_wmma_f32_16x16x4_f32(
            false, a0, false, b0, (short)0, acc0, false, false);
        acc1 = __builtin_amdgcn_wmma_f32_16x16x4_f32(
            false, a1, false, b1, (short)0, acc1, false, false);
    }
    v8f acc = acc0 + acc1;

    const int col = n0 + l15;
#pragma unroll
    for (int i = 0; i < 8; ++i) {
        int row = m0 + half * 8 + i;                    // C/D VGPR layout
        float v = acc[i];
        if (bias) v += bias[col];
        if (fuse_mish) v = mish_f(v);
        C[(size_t)row * N + col] = v;
    }
}

// ---------------------------------------------------------------------------
// Degree / normalization
// ---------------------------------------------------------------------------
__global__ void deg_init(float* __restrict__ deg, int n) {
    int i = blockIdx.x * blockDim.x + threadIdx.x;
    if (i < n) deg[i] = 1.0f;                           // self-loop
}
__global__ void deg_count(const int* __restrict__ col, float* __restrict__ deg, int E) {
    int e = blockIdx.x * blockDim.x + threadIdx.x;
    if (e < E) atomicAdd(&deg[col[e]], 1.0f);
}
__global__ void deg_rsqrt(const float* __restrict__ deg, float* __restrict__ dis, int n) {
    int i = blockIdx.x * blockDim.x + threadIdx.x;
    if (i < n) dis[i] = rsqrtf(deg[i]);
}

// ---------------------------------------------------------------------------
// agg init with self-loop term:  agg[i,:] = xw[i,:] * dis[i]^2   (float4)
// ---------------------------------------------------------------------------
__global__ void self_loop_init(const float* __restrict__ xw,
                               const float* __restrict__ dis,
                               float* __restrict__ agg, int n) {
    size_t i4 = (size_t)blockIdx.x * blockDim.x + threadIdx.x;
    if (i4 < (size_t)n * (FEATS / 4)) {
        int node = (int)(i4 >> 5);                      // (i4*4)/128
        float d = dis[node];
        float s = d * d;
        float4 v = *(const float4*)&xw[i4 * 4];
        v.x *= s; v.y *= s; v.z *= s; v.w *= s;
        *(float4*)&agg[i4 * 4] = v;
    }
}

// edge scatter: one wave per edge; lane L handles feats [4L, 4L+4)
__global__ void edge_scatter(const float* __restrict__ xw,
                             float* __restrict__ agg,
                             const int* __restrict__ row,
                             const int* __restrict__ col,
                             const float* __restrict__ dis, int E) {
    size_t gid = (size_t)blockIdx.x * blockDim.x + threadIdx.x;
    if (gid < (size_t)E * (FEATS / 4)) {
        int e  = (int)(gid >> 5);                       // 32 lanes : 1 edge
        int f4 = (int)(gid & 31) << 2;
        int r = row[e], c = col[e];
        float nrm = dis[r] * dis[c];
        float4 v = *(const float4*)&xw[(size_t)r * FEATS + f4];
        float* dst = &agg[(size_t)c * FEATS + f4];
        atomicAdd(dst + 0, v.x * nrm);
        atomicAdd(dst + 1, v.y * nrm);
        atomicAdd(dst + 2, v.z * nrm);
        atomicAdd(dst + 3, v.w * nrm);
    }
}

// epilogue: h = mish(agg + bias)  (in place, float4)
__global__ void bias_mish(float* __restrict__ h, const float* __restrict__ bias, size_t total4) {
    size_t i4 = (size_t)blockIdx.x * blockDim.x + threadIdx.x;
    if (i4 < total4) {
        int f = (int)((i4 * 4) & (FEATS - 1));
        float4 b = *(const float4*)&bias[f];
        float4 v = *(float4*)&h[i4 * 4];
        v.x = mish_f(v.x + b.x);
        v.y = mish_f(v.y + b.y);
        v.z = mish_f(v.z + b.z);
        v.w = mish_f(v.w + b.w);
        *(float4*)&h[i4 * 4] = v;
    }
}

// ---------------------------------------------------------------------------
// Pooling + BatchNorm
// ---------------------------------------------------------------------------
__global__ void zero_f32(float* __restrict__ p, int n) {
    int i = blockIdx.x * blockDim.x + threadIdx.x;
    if (i < n) p[i] = 0.0f;
}
__global__ void pool_sum(const float* __restrict__ x1, const int* __restrict__ batch,
                         float* __restrict__ psum, int n) {
    size_t gid = (size_t)blockIdx.x * blockDim.x + threadIdx.x;
    if (gid < (size_t)n * (OUTF / 4)) {
        int node = (int)(gid >> 4);                     // 16 float4 per node
        int f4   = (int)(gid & 15) << 2;
        float4 v = *(const float4*)&x1[(size_t)node * OUTF + f4];
        float* dst = &psum[(size_t)batch[node] * OUTF + f4];
        atomicAdd(dst + 0, v.x);
        atomicAdd(dst + 1, v.y);
        atomicAdd(dst + 2, v.z);
        atomicAdd(dst + 3, v.w);
    }
}
__global__ void pool_cnt(const int* __restrict__ batch, float* __restrict__ cnt, int n) {
    int i = blockIdx.x * blockDim.x + threadIdx.x;
    if (i < n) atomicAdd(&cnt[batch[i]], 1.0f);
}
// one block, 64 threads: per-feature batch-norm over 512 pooled rows
__global__ void batchnorm(const float* __restrict__ psum, const float* __restrict__ cnt,
                          const float* __restrict__ gamma, const float* __restrict__ beta,
                          float* __restrict__ out) {
    int f = threadIdx.x;  // 0..63
    float mu = 0.0f;
    for (int g = 0; g < NGRAPH; ++g)
        mu += psum[g * OUTF + f] / fmaxf(cnt[g], 1.0f);
    mu *= (1.0f / NGRAPH);
    float var = 0.0f;
    for (int g = 0; g < NGRAPH; ++g) {
        float p = psum[g * OUTF + f] / fmaxf(cnt[g], 1.0f) - mu;
        var += p * p;
    }
    var *= (1.0f / NGRAPH);
    float inv = rsqrtf(var + BN_EPS);
    for (int g = 0; g < NGRAPH; ++g) {
        float p = psum[g * OUTF + f] / fmaxf(cnt[g], 1.0f);
        out[g * OUTF + f] = (p - mu) * inv * gamma[f] + beta[f];
    }
}

// ---------------------------------------------------------------------------
extern "C" void kernel_launch(void* const* d_in, const int* in_sizes, int n_in,
                              void* d_out, int out_size, void* d_ws, size_t ws_size,
                              hipStream_t stream) {
    const float* x     = (const float*)d_in[0];
    const int*   eidx  = (const int*)d_in[1];
    const int*   batch = (const int*)d_in[2];
    const float* W1    = (const float*)d_in[3];
    const float* b1    = (const float*)d_in[4];
    const float* W2    = (const float*)d_in[5];
    const float* b2    = (const float*)d_in[6];
    const float* Wfc   = (const float*)d_in[7];
    const float* bfc   = (const float*)d_in[8];
    const float* gamma = (const float*)d_in[9];
    const float* beta  = (const float*)d_in[10];

    const int Nn = in_sizes[0] / FEATS;   // 50000 (multiple of 16)
    const int E  = in_sizes[1] / 2;       // 800000
    const int* erow = eidx;
    const int* ecol = eidx + E;

    // workspace layout (floats): two 25.6 MB feature buffers + small scratch
    const size_t SZ = (size_t)Nn * FEATS;
    float* bufA = (float*)d_ws;                  // xw   (ping)
    float* bufB = bufA + SZ;                     // agg/h (pong)
    float* deg  = bufB + SZ;
    float* dis  = deg + Nn;
    float* psum = dis + Nn;                      // 512*64
    float* cnt  = psum + NGRAPH * OUTF;          // 512

    float* out_x1 = (float*)d_out;               // [Nn, 64]
    float* out_x2 = out_x1 + (size_t)Nn * OUTF;  // [512, 64]

    const int T = 256;
    const int gN   = (Nn + T - 1) / T;
    const int gE   = (E + T - 1) / T;
    const int gNF4 = (int)((SZ / 4 + T - 1) / T);
    const int gEF4 = (int)(((size_t)E * (FEATS / 4) + T - 1) / T);
    const int gNO4 = (int)(((size_t)Nn * (OUTF / 4) + T - 1) / T);
    const int mt   = Nn / 16;                    // 3125 row tiles

    // 1) degrees / symmetric norm
    deg_init<<<gN, T, 0, stream>>>(deg, Nn);
    deg_count<<<gE, T, 0, stream>>>(ecol, deg, E);
    deg_rsqrt<<<gN, T, 0, stream>>>(deg, dis, Nn);

    // 2) layer 1: xw = x @ W1 -> A ; agg -> B ; mish(agg+b1) in place
    gemm_wmma_f32<<<mt, 256, 0, stream>>>(x, W1, bufA, nullptr, Nn, FEATS, FEATS, 0);
    self_loop_init<<<gNF4, T, 0, stream>>>(bufA, dis, bufB, Nn);
    edge_scatter<<<gEF4, T, 0, stream>>>(bufA, bufB, erow, ecol, dis, E);
    bias_mish<<<gNF4, T, 0, stream>>>(bufB, b1, SZ / 4);

    // 3) layer 2: xw = h1 @ W2 -> A ; agg -> B (h1 dead) ; mish(agg+b2)
    gemm_wmma_f32<<<mt, 256, 0, stream>>>(bufB, W2, bufA, nullptr, Nn, FEATS, FEATS, 0);
    self_loop_init<<<gNF4, T, 0, stream>>>(bufA, dis, bufB, Nn);
    edge_scatter<<<gEF4, T, 0, stream>>>(bufA, bufB, erow, ecol, dis, E);
    bias_mish<<<gNF4, T, 0, stream>>>(bufB, b2, SZ / 4);

    // 4) x1 = mish(h2 @ Wfc + bfc) -> d_out   (N=64 -> 4 waves/block)
    gemm_wmma_f32<<<mt, 128, 0, stream>>>(bufB, Wfc, out_x1, bfc, Nn, FEATS, OUTF, 1);

    // 5) mean pool + batchnorm -> d_out tail
    zero_f32<<<(NGRAPH * OUTF + NGRAPH + T - 1) / T, T, 0, stream>>>(psum, NGRAPH * OUTF + NGRAPH);
    pool_sum<<<gNO4, T, 0, stream>>>(out_x1, batch, psum, Nn);
    pool_cnt<<<gN, T, 0, stream>>>(batch, cnt, Nn);
    batchnorm<<<1, OUTF, 0, stream>>>(psum, cnt, gamma, beta, out_x2);
}